// SeparateSourceDense_74448963108908
// MI455X (gfx1250) — compile-verified
//
#include <hip/hip_runtime.h>

typedef float v2f __attribute__((ext_vector_type(2)));
typedef float v8f __attribute__((ext_vector_type(8)));

static constexpr int Nn = 8192;   // rows
static constexpr int Ff = 256;    // fan-in
static constexpr int Oo = 256;    // fan-out
static constexpr int Ss = 16;     // sources
static constexpr int MT = 64;     // M rows per block (4 subtiles of 16)

// ---------------------------------------------------------------------------
// Phase 1: counting sort of row indices by source -> buckets in workspace.
//   d_ws layout: offsets[0..16] (17 ints, padded to 32), then row_idx[8192].
// ---------------------------------------------------------------------------
__global__ __launch_bounds__(256) void ssd_bucket_kernel(
    const int* __restrict__ source, int* __restrict__ offsets,
    int* __restrict__ row_idx) {
  __shared__ int hist[Ss];
  __shared__ int cursor[Ss];
  const int t = threadIdx.x;
  if (t < Ss) hist[t] = 0;
  __syncthreads();
  for (int n = t; n < Nn; n += blockDim.x) {
    int s = source[n] & (Ss - 1);
    atomicAdd(&hist[s], 1);
  }
  __syncthreads();
  if (t == 0) {
    int run = 0;
    for (int s = 0; s < Ss; ++s) {
      cursor[s] = run;
      offsets[s] = run;
      run += hist[s];
    }
    offsets[Ss] = run;
  }
  __syncthreads();
  for (int n = t; n < Nn; n += blockDim.x) {
    int s = source[n] & (Ss - 1);
    int pos = atomicAdd(&cursor[s], 1);
    row_idx[pos] = n;
  }
}

// ---------------------------------------------------------------------------
// Phase 2: grouped GEMM over fp32 WMMA (V_WMMA_F32_16X16X4_F32).
// Block = 64 bucketed rows x 256 cols. 8 waves; wave w owns columns
// [32w, 32w+32) as two 16x16 col-tiles, accumulated for 4 M-subtiles,
// so each B fragment is reused by 4 WMMAs (8 VMEM : 8 WMMA per K=4 step).
// ---------------------------------------------------------------------------
__global__ __launch_bounds__(256) void ssd_gemm_kernel(
    const float* __restrict__ x, const float* __restrict__ kernelw,
    const float* __restrict__ bias, const int* __restrict__ offsets,
    const int* __restrict__ row_idx, float* __restrict__ out) {
  const int s    = blockIdx.y;              // source id
  const int tile = blockIdx.x;              // 64-row tile within this bucket
  const int beg = offsets[s];
  const int end = offsets[s + 1];
  const int mstart = beg + tile * MT;
  if (mstart >= end) return;                // over-provisioned grid: idle

  const int lane = threadIdx.x & 31;
  const int wave = threadIdx.x >> 5;
  const int half = lane >> 4;               // 0: lanes 0-15, 1: lanes 16-31
  const int l15  = lane & 15;

  // Per-subtile A row pointers. A layout (32-bit A 16x4): lane l carries
  // row M = l&15 in both halves; K = 2*half + {0,1} -> one b64 per step.
  const float* __restrict__ xrow[4];
#pragma unroll
  for (int sub = 0; sub < 4; ++sub) {
    int aIdx = mstart + sub * 16 + l15;
    if (aIdx > end - 1) aIdx = end - 1;     // clamp gather for partial tiles
    xrow[sub] = x + (size_t)row_idx[aIdx] * Ff;
  }

  const float* __restrict__ W = kernelw + (size_t)s * Ff * Oo;
  const int col0 = wave * 32 + l15;         // N for col-tile 0 (this lane)
  const int col1 = col0 + 16;               // N for col-tile 1

  v8f acc[4][2];
#pragma unroll
  for (int sub = 0; sub < 4; ++sub) {
    acc[sub][0] = v8f{0.f, 0.f, 0.f, 0.f, 0.f, 0.f, 0.f, 0.f};
    acc[sub][1] = v8f{0.f, 0.f, 0.f, 0.f, 0.f, 0.f, 0.f, 0.f};
  }

#pragma unroll 2
  for (int k = 0; k < Ff; k += 4) {
    const int ka = k + 2 * half;
    // B fragments (shared across the 4 M-subtiles):
    // lane l: N = l&15, VGPR{0,1} = K{2*half, 2*half+1}.
    v2f b0, b1;
    b0.x = W[(size_t)(ka + 0) * Oo + col0];
    b0.y = W[(size_t)(ka + 1) * Oo + col0];
    b1.x = W[(size_t)(ka + 0) * Oo + col1];
    b1.y = W[(size_t)(ka + 1) * Oo + col1];
    // A fragments: one coalesced b64 per subtile.
    v2f a[4];
#pragma unroll
    for (int sub = 0; sub < 4; ++sub) a[sub] = *(const v2f*)(xrow[sub] + ka);
#pragma unroll
    for (int sub = 0; sub < 4; ++sub) {
      acc[sub][0] = __builtin_amdgcn_wmma_f32_16x16x4_f32(
          false, a[sub], false, b0, (short)0, acc[sub][0], false, false);
      acc[sub][1] = __builtin_amdgcn_wmma_f32_16x16x4_f32(
          false, a[sub], false, b1, (short)0, acc[sub][1], false, false);
    }
  }

  // Bias: per C/D layout, every accumulator element in this lane shares N.
  const float bias0 = bias[s * Oo + col0];
  const float bias1 = bias[s * Oo + col1];

  // C/D layout: VGPR r holds M = r + 8*half, N = l&15.
#pragma unroll
  for (int sub = 0; sub < 4; ++sub) {
#pragma unroll
    for (int r = 0; r < 8; ++r) {
      const int mi = mstart + sub * 16 + r + 8 * half;
      if (mi < end) {
        const size_t orow = (size_t)row_idx[mi];
        out[orow * Oo + col0] = acc[sub][0][r] + bias0;
        out[orow * Oo + col1] = acc[sub][1][r] + bias1;
      }
    }
  }
}

// ---------------------------------------------------------------------------
// Launch: inputs in setup_inputs() order: x(f32), source(int), kernel(f32),
// bias_param(f32). Output: f32 [8192, 256].
// ---------------------------------------------------------------------------
extern "C" void kernel_launch(void* const* d_in, const int* in_sizes, int n_in,
                              void* d_out, int out_size, void* d_ws, size_t ws_size,
                              hipStream_t stream) {
  const float* x       = (const float*)d_in[0];
  const int*   source  = (const int*)d_in[1];
  const float* kernelw = (const float*)d_in[2];
  const float* bias    = (const float*)d_in[3];
  float* out = (float*)d_out;

  int* offsets = (int*)d_ws;        // 17 ints used, 32 reserved
  int* row_idx = offsets + 32;      // 8192 ints

  ssd_bucket_kernel<<<1, 256, 0, stream>>>(source, offsets, row_idx);

  dim3 grid(Nn / MT, Ss, 1);        // over-provisioned: idle tiles early-exit
  ssd_gemm_kernel<<<grid, 256, 0, stream>>>(x, kernelw, bias, offsets,
                                            row_idx, out);
}